// DoubleGlobalAttention_81063212745411
// MI455X (gfx1250) — compile-verified
//
#include <hip/hip_runtime.h>
#include <hip/hip_bf16.h>
#include <math.h>

// ---------------------------------------------------------------------------
// DoubleGlobalAttention for MI455X (gfx1250, wave32, WMMA)
// B=64, D=1024, L=2048, Dh=512
// ---------------------------------------------------------------------------

typedef float v2f __attribute__((ext_vector_type(2)));
typedef float v8f __attribute__((ext_vector_type(8)));

#define PF 4   // software-pipeline depth (stages of k-step fragments)

// ---------------------------------------------------------------------------
// GEMM: C[64 x N] = A[64 x K] * Bt[N x K]^T   (optional tanh epilogue)
// Block = 128 threads = 4 waves. Each wave owns a K/4 slice and computes the
// full 64x16 strip with four v_wmma_f32_16x16x4_f32 accumulators; partial
// results are reduced across waves in LDS.  Fragment loads are software-
// pipelined PF stages deep; the steady-state loop is branch-free (always
// refills) with the final block peeled, so loads stay clustered in clauses
// and s_wait_loadcnt never drains to zero before a WMMA.
//
// f32 WMMA VGPR layouts (ISA 7.12.2):
//   A 16x4 : lane l<16 -> M=l (v0=K0,v1=K1); lane>=16 -> M=l-16 (K2,K3)
//   B 4x16 : lane l<16 -> N=l (v0=K0,v1=K1); lane>=16 -> N=l-16 (K2,K3)
//   C 16x16: VGPR r = row M=r (lanes 0-15) / M=r+8 (lanes 16-31), N = lane&15
// ---------------------------------------------------------------------------
__global__ __launch_bounds__(128) void wmma_gemm_m64(
    const float* __restrict__ A,   // [64 x K] row-major
    const float* __restrict__ Bt,  // [N x K] row-major (torch Linear weight)
    float* __restrict__ C,         // [64 x N]
    int N, int K, int applyTanh)
{
    const int tileN = blockIdx.x * 16;
    const int tid   = threadIdx.x;
    const int lane  = tid & 31;
    const int wave  = tid >> 5;        // 0..3 : K slice
    const int hi    = lane >> 4;       // 0: K0/K1, 1: K2/K3
    const int l15   = lane & 15;

    const int Kslice = K >> 2;         // per-wave K extent (256 or 512)
    const int kBase  = wave * Kslice + hi * 2;
    const int nblk   = Kslice / (4 * PF);

    __shared__ float redbuf[4 * 1024]; // 16 KB: [wave][tile*256 + r*32 + lane]

    v8f acc0 = {}, acc1 = {}, acc2 = {}, acc3 = {};

    const float* a0p = A + (size_t)(l15 +  0) * K + kBase;
    const float* a1p = A + (size_t)(l15 + 16) * K + kBase;
    const float* a2p = A + (size_t)(l15 + 32) * K + kBase;
    const float* a3p = A + (size_t)(l15 + 48) * K + kBase;
    const float* bp  = Bt + (size_t)(tileN + l15) * K + kBase;

    // ---- prologue: fill PF pipeline stages ----
    v2f bb[PF], fa0[PF], fa1[PF], fa2[PF], fa3[PF];
#pragma unroll
    for (int s = 0; s < PF; ++s) {
        const int k0 = 4 * s;
        bb[s].x  = bp[k0];  bb[s].y  = bp[k0 + 1];
        fa0[s].x = a0p[k0]; fa0[s].y = a0p[k0 + 1];
        fa1[s].x = a1p[k0]; fa1[s].y = a1p[k0 + 1];
        fa2[s].x = a2p[k0]; fa2[s].y = a2p[k0 + 1];
        fa3[s].x = a3p[k0]; fa3[s].y = a3p[k0 + 1];
    }

    // ---- steady state: branch-free consume + refill (immediate offsets) ----
    for (int blk = 0; blk < nblk - 1; ++blk) {
#pragma unroll
        for (int s = 0; s < PF; ++s) {
            acc0 = __builtin_amdgcn_wmma_f32_16x16x4_f32(false, fa0[s], false, bb[s],
                                                         (short)0, acc0, false, false);
            acc1 = __builtin_amdgcn_wmma_f32_16x16x4_f32(false, fa1[s], false, bb[s],
                                                         (short)0, acc1, false, false);
            acc2 = __builtin_amdgcn_wmma_f32_16x16x4_f32(false, fa2[s], false, bb[s],
                                                         (short)0, acc2, false, false);
            acc3 = __builtin_amdgcn_wmma_f32_16x16x4_f32(false, fa3[s], false, bb[s],
                                                         (short)0, acc3, false, false);
            const int k0 = 4 * PF + 4 * s;   // next block, same stage
            bb[s].x  = bp[k0];  bb[s].y  = bp[k0 + 1];
            fa0[s].x = a0p[k0]; fa0[s].y = a0p[k0 + 1];
            fa1[s].x = a1p[k0]; fa1[s].y = a1p[k0 + 1];
            fa2[s].x = a2p[k0]; fa2[s].y = a2p[k0 + 1];
            fa3[s].x = a3p[k0]; fa3[s].y = a3p[k0 + 1];
        }
        bp  += 4 * PF;
        a0p += 4 * PF;
        a1p += 4 * PF;
        a2p += 4 * PF;
        a3p += 4 * PF;
    }

    // ---- epilogue: consume final PF stages, no refill ----
#pragma unroll
    for (int s = 0; s < PF; ++s) {
        acc0 = __builtin_amdgcn_wmma_f32_16x16x4_f32(false, fa0[s], false, bb[s],
                                                     (short)0, acc0, false, false);
        acc1 = __builtin_amdgcn_wmma_f32_16x16x4_f32(false, fa1[s], false, bb[s],
                                                     (short)0, acc1, false, false);
        acc2 = __builtin_amdgcn_wmma_f32_16x16x4_f32(false, fa2[s], false, bb[s],
                                                     (short)0, acc2, false, false);
        acc3 = __builtin_amdgcn_wmma_f32_16x16x4_f32(false, fa3[s], false, bb[s],
                                                     (short)0, acc3, false, false);
    }

    // ---- cross-wave K reduction in LDS ----
    float* my = redbuf + wave * 1024;
#pragma unroll
    for (int r = 0; r < 8; ++r) {
        my[0 * 256 + r * 32 + lane] = acc0[r];
        my[1 * 256 + r * 32 + lane] = acc1[r];
        my[2 * 256 + r * 32 + lane] = acc2[r];
        my[3 * 256 + r * 32 + lane] = acc3[r];
    }
    __syncthreads();

    for (int idx = tid; idx < 1024; idx += 128) {
        float v = redbuf[idx] + redbuf[1024 + idx] +
                  redbuf[2048 + idx] + redbuf[3072 + idx];
        if (applyTanh) v = tanhf(v);
        const int t  = idx >> 8;          // which 16-row tile
        const int r  = (idx >> 5) & 7;    // C VGPR index
        const int ln = idx & 31;          // lane
        const int m  = t * 16 + r + ((ln >= 16) ? 8 : 0);
        const int n  = tileN + (ln & 15);
        C[(size_t)m * N + n] = v;
    }
}

// ---------------------------------------------------------------------------
// Wave / block reductions (wave32)
// ---------------------------------------------------------------------------
__device__ __forceinline__ float waveReduceSum(float v) {
#pragma unroll
    for (int off = 16; off > 0; off >>= 1) v += __shfl_xor(v, off, 32);
    return v;
}
__device__ __forceinline__ float waveReduceMax(float v) {
#pragma unroll
    for (int off = 16; off > 0; off >>= 1) v = fmaxf(v, __shfl_xor(v, off, 32));
    return v;
}
__device__ __forceinline__ float blockReduceSum(float v, float* red, int tid) {
    const int lane = tid & 31, wave = tid >> 5;
    v = waveReduceSum(v);
    if (lane == 0) red[wave] = v;
    __syncthreads();
    if (wave == 0) {
        float x = (lane < 16) ? red[lane] : 0.0f;
        x = waveReduceSum(x);
        if (lane == 0) red[0] = x;
    }
    __syncthreads();
    float r = red[0];
    __syncthreads();
    return r;
}
__device__ __forceinline__ float blockReduceMax(float v, float* red, int tid) {
    const int lane = tid & 31, wave = tid >> 5;
    v = waveReduceMax(v);
    if (lane == 0) red[wave] = v;
    __syncthreads();
    if (wave == 0) {
        float x = (lane < 16) ? red[lane] : -INFINITY;
        x = waveReduceMax(x);
        if (lane == 0) red[0] = x;
    }
    __syncthreads();
    float r = red[0];
    __syncthreads();
    return r;
}

// ---------------------------------------------------------------------------
// Fused dual-query attention: one workgroup (512 threads = 16 waves) per batch.
// All streaming loads are b128 (float4). Scores + softmax live entirely in LDS;
// normalized probabilities go straight to d_out attn regions.
// ---------------------------------------------------------------------------
__global__ __launch_bounds__(512) void attn_fused(
    const float* __restrict__ ctx,   // [64, 2048, 512]
    const float* __restrict__ Q,     // [64, 1024] interleaved (q0,q1) pairs
    const float* __restrict__ inp,   // [64, 1024]
    float* __restrict__ attn0,       // [64, 2048]
    float* __restrict__ attn1,       // [64, 2048]
    float* __restrict__ combined)    // [64, 2048]
{
    const int b    = blockIdx.x;
    const int tid  = threadIdx.x;     // 0..511
    const int lane = tid & 31;
    const int wave = tid >> 5;        // 0..15

    __shared__ __align__(16) float q0s[512];
    __shared__ __align__(16) float q1s[512];
    __shared__ __align__(16) float s0s[2048];
    __shared__ __align__(16) float s1s[2048];
    __shared__ float red[16];

    // targetT reshape(B, D/2, 2): q_k[d] = Q[b][2d + k]
    const float* qb = Q + (size_t)b * 1024;
    q0s[tid] = qb[2 * tid];
    q1s[tid] = qb[2 * tid + 1];
    __syncthreads();

    const float*  cb  = ctx + (size_t)b * 2048 * 512;
    const float4* cb4 = (const float4*)cb;              // row stride = 128 f4
    const float4* q04 = (const float4*)q0s;
    const float4* q14 = (const float4*)q1s;

    // ---- Pass 1: scores s_k[l] = ctx[b,l,:] . q_k  (one wave per row) ----
    for (int row = wave; row < 2048; row += 16) {
        const float4* cr4 = cb4 + (size_t)row * 128;
        float a0 = 0.0f, a1 = 0.0f;
#pragma unroll
        for (int i = 0; i < 4; ++i) {                   // 128 f4 / 32 lanes
            const int d4 = lane + 32 * i;
            const float4 c  = cr4[d4];
            const float4 u0 = q04[d4];
            const float4 u1 = q14[d4];
            a0 = fmaf(c.x, u0.x, a0); a0 = fmaf(c.y, u0.y, a0);
            a0 = fmaf(c.z, u0.z, a0); a0 = fmaf(c.w, u0.w, a0);
            a1 = fmaf(c.x, u1.x, a1); a1 = fmaf(c.y, u1.y, a1);
            a1 = fmaf(c.z, u1.z, a1); a1 = fmaf(c.w, u1.w, a1);
        }
        a0 = waveReduceSum(a0);
        a1 = waveReduceSum(a1);
        if (lane == 0) { s0s[row] = a0; s1s[row] = a1; }
    }
    __syncthreads();

    // ---- Softmax over L=2048, entirely in LDS ----
    float m0 = -INFINITY, m1 = -INFINITY;
    for (int l = tid; l < 2048; l += 512) {
        m0 = fmaxf(m0, s0s[l]);
        m1 = fmaxf(m1, s1s[l]);
    }
    const float M0 = blockReduceMax(m0, red, tid);
    const float M1 = blockReduceMax(m1, red, tid);

    float e0sum = 0.0f, e1sum = 0.0f;
    for (int l = tid; l < 2048; l += 512) {
        const float e0 = __expf(s0s[l] - M0);
        const float e1 = __expf(s1s[l] - M1);
        s0s[l] = e0; e0sum += e0;
        s1s[l] = e1; e1sum += e1;
    }
    const float S0 = blockReduceSum(e0sum, red, tid);
    const float S1 = blockReduceSum(e1sum, red, tid);
    const float inv0 = 1.0f / S0;
    const float inv1 = 1.0f / S1;

    float* a0out = attn0 + (size_t)b * 2048;
    float* a1out = attn1 + (size_t)b * 2048;
    for (int l = tid; l < 2048; l += 512) {
        const float p0 = s0s[l] * inv0;
        const float p1 = s1s[l] * inv1;
        s0s[l] = p0; a0out[l] = p0;
        s1s[l] = p1; a1out[l] = p1;
    }
    __syncthreads();

    // ---- Pass 2: weighted[d] = sum_l p[l]*ctx[b,l,d] ----
    // thread = (row-slice rs of 512 rows) x (column group cg of 4 columns)
    const int cg = tid & 127;          // columns 4*cg .. 4*cg+3
    const int rs = tid >> 7;           // 0..3
    float4 w0 = {0, 0, 0, 0}, w1 = {0, 0, 0, 0};
    const int l0 = rs * 512;
#pragma unroll 4
    for (int l = l0; l < l0 + 512; ++l) {
        const float4 c  = cb4[(size_t)l * 128 + cg];    // coalesced b128
        const float  p0 = s0s[l];                       // LDS broadcast
        const float  p1 = s1s[l];
        w0.x = fmaf(p0, c.x, w0.x); w0.y = fmaf(p0, c.y, w0.y);
        w0.z = fmaf(p0, c.z, w0.z); w0.w = fmaf(p0, c.w, w0.w);
        w1.x = fmaf(p1, c.x, w1.x); w1.y = fmaf(p1, c.y, w1.y);
        w1.z = fmaf(p1, c.z, w1.z); w1.w = fmaf(p1, c.w, w1.w);
    }
    __syncthreads();                    // p values no longer needed

    // reduce the 4 row-slices in LDS (reuse score buffers)
    ((float4*)s0s)[rs * 128 + cg] = w0; // s0s[rs*512 + col]
    ((float4*)s1s)[rs * 128 + cg] = w1;
    __syncthreads();

    const float w0f = s0s[tid] + s0s[512 + tid] + s0s[1024 + tid] + s0s[1536 + tid];
    const float w1f = s1s[tid] + s1s[512 + tid] + s1s[1024 + tid] + s1s[1536 + tid];

    // combined[b] = [w0(512) | w1(512) | input(1024)]
    float* comb = combined + (size_t)b * 2048;
    comb[tid]       = w0f;
    comb[512 + tid] = w1f;
    const float* ib = inp + (size_t)b * 1024;
    comb[1024 + tid] = ib[tid];
    comb[1536 + tid] = ib[512 + tid];
}

// ---------------------------------------------------------------------------
extern "C" void kernel_launch(void* const* d_in, const int* in_sizes, int n_in,
                              void* d_out, int out_size, void* d_ws, size_t ws_size,
                              hipStream_t stream) {
    (void)in_sizes; (void)n_in; (void)out_size; (void)ws_size;

    const float* input   = (const float*)d_in[0];   // [64,1024]
    const float* context = (const float*)d_in[1];   // [64,2048,512]
    const float* W_in    = (const float*)d_in[2];   // [1024,1024]
    const float* W_out   = (const float*)d_in[3];   // [1024,2048]

    float* out    = (float*)d_out;
    float* ctxOut = out;                       // [64,1024]
    float* attn0  = out + 64 * 1024;           // [64,2048]
    float* attn1  = attn0 + 64 * 2048;         // [64,2048]

    float* Q        = (float*)d_ws;            // [64,1024]
    float* combined = Q + 64 * 1024;           // [64,2048]

    // 1) Q = input @ W_in^T    (M=64, N=1024, K=1024)
    wmma_gemm_m64<<<dim3(1024 / 16), dim3(128), 0, stream>>>(
        input, W_in, Q, 1024, 1024, /*tanh=*/0);

    // 2) fused dual-query attention per batch
    attn_fused<<<dim3(64), dim3(512), 0, stream>>>(
        context, Q, input, attn0, attn1, combined);

    // 3) out = tanh(combined @ W_out^T)   (M=64, N=1024, K=2048)
    wmma_gemm_m64<<<dim3(1024 / 16), dim3(128), 0, stream>>>(
        combined, W_out, ctxOut, 1024, 2048, /*tanh=*/1);
}